// BSplineParameterization_88734024336030
// MI455X (gfx1250) — compile-verified
//
#include <hip/hip_runtime.h>

#define T_PTS  262144
#define S_COLS 512
#define K_ROWS 258      // NUM_KNOTS + DEGREE - 1
#define NKNOTS 262      // NUM_KNOTS + 2*DEGREE
#define DEG    3

typedef float v2f __attribute__((ext_vector_type(2)));
typedef float v8f __attribute__((ext_vector_type(8)));

__global__ __launch_bounds__(256)
void bspline_wmma_kernel(const float* __restrict__ times,
                         const float* __restrict__ weights,
                         const float* __restrict__ knots,
                         float* __restrict__ bsp,     // T x S
                         float* __restrict__ sums)    // T
{
    __shared__ float sK[NKNOTS];     // knot vector
    __shared__ float sA[16][16];     // dense basis tile (16 times x 16-row window)
    __shared__ int   sMi[16];        // interval index per time
    __shared__ int   sBase;          // window base row
    __shared__ float sSum[16];       // per-time row sums

    const int tid = threadIdx.x;
    const int tb  = blockIdx.x;      // block of 16 consecutive (sorted) times

    for (int i = tid; i < NKNOTS; i += 256) sK[i] = knots[i];
    sA[tid >> 4][tid & 15] = 0.0f;
    if (tid < 16) sSum[tid] = 0.0f;
    __syncthreads();

    // --- threads 0..15: de Boor cubic basis (4 nonzeros) for time r = tid ---
    float Nv0 = 0.0f, Nv1 = 0.0f, Nv2 = 0.0f, Nv3 = 0.0f;
    if (tid < 16) {
        float t  = times[tb * 16 + tid];
        float t0 = sK[DEG];                 // first interior knot == times[0]
        float tl = sK[NKNOTS - DEG - 1];    // last interior knot  == times[-1]
        float hh = (tl - t0) * (1.0f / 255.0f);
        int m = (int)floorf((t - t0) / hh);
        m = min(max(m, 0), 254);
        // exact fixup against the real (linspace-rounded) knots
        while (m > 0   && t <  sK[DEG + m])     --m;
        while (m < 254 && t >= sK[DEG + m + 1]) ++m;
        sMi[tid] = m;
        if (t < tl) {   // t == t_last -> all-zero row (half-open intervals)
            int ell = DEG + m;
            float N[4]; N[0] = 1.0f;
            float left[4], right[4];
            for (int d = 1; d <= DEG; ++d) {
                left[d]  = t - sK[ell + 1 - d];
                right[d] = sK[ell + d] - t;
                float saved = 0.0f;
                for (int r2 = 0; r2 < d; ++r2) {
                    float den = right[r2 + 1] + left[d - r2];
                    float tmp = (den != 0.0f) ? (N[r2] / den) : 0.0f;   // drop zero-denominator terms
                    N[r2] = saved + right[r2 + 1] * tmp;
                    saved = left[d - r2] * tmp;
                }
                N[d] = saved;
            }
            Nv0 = N[0]; Nv1 = N[1]; Nv2 = N[2]; Nv3 = N[3];
        }
    }
    __syncthreads();

    if (tid == 0) {
        int m0 = sMi[0];
        for (int i = 1; i < 16; ++i) m0 = min(m0, sMi[i]);
        sBase = min(m0, K_ROWS - 16);   // 16-row weight window stays in bounds
    }
    __syncthreads();

    if (tid < 16) {
        int c = sMi[tid] - sBase;       // column in window for basis index m
        c = min(max(c, 0), 12);
        sA[tid][c    ] = Nv0;
        sA[tid][c + 1] = Nv1;
        sA[tid][c + 2] = Nv2;
        sA[tid][c + 3] = Nv3;
    }
    __syncthreads();

    // --- WMMA phase: 8 waves x 4 column tiles = 512 output columns ---
    const int lane = tid & 31;
    const int wv   = tid >> 5;       // wave id 0..7
    const int r    = lane & 15;      // A: M row / B,C,D: N column
    const int hsel = lane >> 4;      // half-wave select
    const int base = sBase;

    float rowSum[8];
    #pragma unroll
    for (int v = 0; v < 8; ++v) rowSum[v] = 0.0f;

    for (int q = 0; q < 4; ++q) {
        const int col0 = (wv * 4 + q) * 16;
        const float* Wp = weights + (size_t)base * S_COLS + col0 + r;
        v8f acc = {};
        #pragma unroll
        for (int s = 0; s < 4; ++s) {
            v2f a, b;
            // A lane layout: VGPR v holds K = 4*s + 2*hsel + v, row M = r
            a.x = sA[r][4 * s + 2 * hsel];
            a.y = sA[r][4 * s + 2 * hsel + 1];
            // B lane layout: VGPR v holds K = 4*s + 2*hsel + v, col N = r
            b.x = Wp[(size_t)(4 * s + 2 * hsel)     * S_COLS];
            b.y = Wp[(size_t)(4 * s + 2 * hsel + 1) * S_COLS];
            acc = __builtin_amdgcn_wmma_f32_16x16x4_f32(
                false, a, false, b, (short)0, acc, false, false);
        }
        // D lane layout: VGPR v holds row M = v + 8*hsel, col N = r
        #pragma unroll
        for (int v = 0; v < 8; ++v) {
            int row = v + 8 * hsel;
            bsp[(size_t)(tb * 16 + row) * S_COLS + col0 + r] = acc[v];
            rowSum[v] += acc[v];
        }
    }

    // reduce row sums over the 16 N-lanes within each half-wave
    #pragma unroll
    for (int mask = 1; mask < 16; mask <<= 1) {
        #pragma unroll
        for (int v = 0; v < 8; ++v)
            rowSum[v] += __shfl_xor(rowSum[v], mask, 32);
    }
    if (r == 0) {
        #pragma unroll
        for (int v = 0; v < 8; ++v)
            atomicAdd(&sSum[v + 8 * hsel], rowSum[v]);   // ds_add_f32
    }
    __syncthreads();
    if (tid < 16) sums[tb * 16 + tid] = sSum[tid];
}

extern "C" void kernel_launch(void* const* d_in, const int* in_sizes, int n_in,
                              void* d_out, int out_size, void* d_ws, size_t ws_size,
                              hipStream_t stream) {
    const float* times   = (const float*)d_in[0];
    const float* weights = (const float*)d_in[1];
    const float* knots   = (const float*)d_in[2];
    float* bsp  = (float*)d_out;
    float* sums = bsp + (size_t)T_PTS * S_COLS;

    dim3 grid(T_PTS / 16);
    bspline_wmma_kernel<<<grid, 256, 0, stream>>>(times, weights, knots, bsp, sums);
    (void)in_sizes; (void)n_in; (void)out_size; (void)d_ws; (void)ws_size;
}